// FastPooling_24558622999099
// MI455X (gfx1250) — compile-verified
//
#include <hip/hip_runtime.h>

// Problem constants (from reference): n=512, HIDDEN=128, N_CELLS=4, POOL=8
#define HIDDENF   128
#define OUTF      128
#define GRID_SIDE 32
#define NFINE     (GRID_SIDE * GRID_SIDE)   // 1024 fine cells
#define KDIM      2048                       // HIDDEN * N_CELLS * N_CELLS

typedef float v2f __attribute__((ext_vector_type(2)));
typedef float v8f __attribute__((ext_vector_type(8)));

// ---------------------------------------------------------------------------
// Kernel 1: fused scatter (last-write-wins == atomicMax over ascending j)
//           + 8x8 block-sum pooling. One workgroup per pedestrian i.
//           Winner table lives entirely in LDS (4 KB of the 320 KB/WGP).
// pooled layout matches reference: pooled[i][h*16 + cx*4 + cy]
// ---------------------------------------------------------------------------
__global__ __launch_bounds__(256) void scatter_pool_kernel(
    const float* __restrict__ hidden,   // (n, 128)
    const float* __restrict__ obs2,     // (n, 2)
    float* __restrict__ pooled,         // (n, 2048)
    int n)
{
    __shared__ int winner[NFINE];
    const int i   = blockIdx.x;
    const int tid = threadIdx.x;

    for (int c = tid; c < NFINE; c += 256) winner[c] = -1;
    __syncthreads();

    const float xi = obs2[2 * i + 0];
    const float yi = obs2[2 * i + 1];

    // rel / (CELL_SIDE/POOL) + GRID/2  ==  rel * 4 + 16
    for (int j = tid; j < n; j += 256) {
        if (j == i) continue;
        const float ox = (obs2[2 * j + 0] - xi) * 4.0f + 16.0f;
        const float oy = (obs2[2 * j + 1] - yi) * 4.0f + 16.0f;
        if (ox >= 0.0f && ox < 32.0f && oy >= 0.0f && oy < 32.0f) {
            const int cell = (int)ox * GRID_SIDE + (int)oy;
            atomicMax(&winner[cell], j);   // ascending-j last-write-wins
        }
    }
    __syncthreads();

    // 256 threads = 2 groups of 128; each group owns one coarse cell at a
    // time and accumulates all 128 features in registers. Winner reads are
    // wave-uniform LDS broadcasts; hidden reads are fully coalesced.
    const int h   = tid & (HIDDENF - 1);  // feature index 0..127
    const int grp = tid >> 7;             // 0 or 1
    for (int ccp = 0; ccp < 8; ++ccp) {
        const int cc = ccp * 2 + grp;     // coarse cell 0..15
        const int cx = cc >> 2;
        const int cy = cc & 3;
        float acc = 0.0f;
        #pragma unroll 8
        for (int f = 0; f < 64; ++f) {
            const int px   = f >> 3;
            const int py   = f & 7;
            const int fine = (cx * 8 + px) * GRID_SIDE + (cy * 8 + py);
            const int jw   = winner[fine];
            if (jw >= 0) acc += hidden[jw * HIDDENF + h];
        }
        pooled[i * KDIM + h * 16 + cc] = acc;
    }
}

// ---------------------------------------------------------------------------
// Kernel 2: C(512x128) = relu(P(512x2048) x W(2048x128) + b) using
//           V_WMMA_F32_16X16X4_F32. One wave32 per 16x16 C tile.
//
// A 16x4 (MxK) per-lane layout: lanes 0-15 -> K={0,1}, lanes 16-31 -> K={2,3}
//   => one contiguous float2 (b64) load per lane per step.
// B 4x16 (KxN) mirrored with N across lanes.
// C/D: VGPR r holds M = r + 8*(lane>=16), N = lane&15.
// ---------------------------------------------------------------------------
__global__ __launch_bounds__(128) void gemm_wmma_kernel(
    const float* __restrict__ P,     // (M, 2048)
    const float* __restrict__ Wm,    // (2048, 128)
    const float* __restrict__ bias,  // (128,)
    float* __restrict__ out,         // (M, 128)
    int mtiles)
{
    const int wave = threadIdx.x >> 5;
    const int lane = threadIdx.x & 31;
    const int tile = blockIdx.x * 4 + wave;
    const int mt   = tile >> 3;          // 0..mtiles-1
    const int nt   = tile & 7;           // 0..7
    if (mt >= mtiles) return;

    const int ml = lane & 15;            // M (for A) / N (for B) within tile
    const int hf = lane >> 4;            // 0: K pair {0,1}, 1: K pair {2,3}

    const float* arow = P  + (size_t)(mt * 16 + ml) * KDIM + 2 * hf;
    const float* bcol = Wm + (size_t)(2 * hf) * OUTF + nt * 16 + ml;

    v8f c = {};
    #pragma unroll 4
    for (int k = 0; k < KDIM; k += 4) {
        v2f a = *(const v2f*)(arow + k);             // K = k+2*hf, k+2*hf+1
        v2f b;
        b.x = bcol[(size_t)k * OUTF];                // K = k+2*hf
        b.y = bcol[(size_t)k * OUTF + OUTF];         // K = k+2*hf+1
        // (neg_a, A, neg_b, B, c_mod, C, reuse_a, reuse_b)
        c = __builtin_amdgcn_wmma_f32_16x16x4_f32(
                false, a, false, b, (short)0, c, false, false);
    }

    const int col = nt * 16 + ml;
    const float bb = bias[col];
    #pragma unroll
    for (int r = 0; r < 8; ++r) {
        const int row = mt * 16 + hf * 8 + r;
        float v = c[r] + bb;
        out[(size_t)row * OUTF + col] = v > 0.0f ? v : 0.0f;
    }
}

// ---------------------------------------------------------------------------
extern "C" void kernel_launch(void* const* d_in, const int* in_sizes, int n_in,
                              void* d_out, int out_size, void* d_ws, size_t ws_size,
                              hipStream_t stream)
{
    // setup_inputs order: hidden_state, obs1 (unused), obs2, W, b
    const float* hidden = (const float*)d_in[0];
    const float* obs2   = (const float*)d_in[2];
    const float* Wm     = (const float*)d_in[3];
    const float* bias   = (const float*)d_in[4];
    float*       out    = (float*)d_out;

    const int n = in_sizes[2] / 2;       // 512
    float* pooled = (float*)d_ws;        // (n, 2048) fp32 = 4 MB scratch

    scatter_pool_kernel<<<n, 256, 0, stream>>>(hidden, obs2, pooled, n);

    const int mtiles = n / 16;                 // 32
    const int tiles  = mtiles * (OUTF / 16);   // 256 waves
    gemm_wmma_kernel<<<tiles / 4, 128, 0, stream>>>(pooled, Wm, bias, out, mtiles);
}